// DeformConv_13615046329033
// MI455X (gfx1250) — compile-verified
//
#include <hip/hip_runtime.h>

typedef __attribute__((ext_vector_type(2))) float v2f;
typedef __attribute__((ext_vector_type(8))) float v8f;

#define C_IN   64
#define H_IN   128
#define W_IN   128
#define OC     64
#define NTAP   9            // 3x3 taps
#define KTOT   576          // C_IN * NTAP
#define HW     (H_IN * W_IN)
#define TILE_P 64           // pixels per workgroup (contiguous, within one image)
#define CPC    4            // channels per K-chunk
#define KC     (CPC * NTAP) // 36 K per chunk
#define KP     (KC / 2)     // 18 K-pair rows per chunk
#define P2F    160          // floats per pair-row (80 float2); 32-bank shift per row

__global__ __launch_bounds__(256) void deform_conv_wmma_f32(
    const float* __restrict__ x,       // (4, 64, 128, 128)
    const float* __restrict__ offset,  // (4, 18, 128, 128)
    const float* __restrict__ weight,  // (64, 64, 3, 3)
    float* __restrict__ out)           // (4, 64, 128, 128)
{
    // K-pair layout: element (k, col) -> [(k>>1)*P2F + col*2 + (k&1)]
    // => each WMMA A/B fragment (k, k+1 per lane) is one aligned ds_load_b64.
    __shared__ float sW[KP * P2F];              // weight chunk   (11.25 KB)
    __shared__ float sS[KP * P2F];              // im2col chunk   (11.25 KB)
    __shared__ int   sIdx[NTAP * TILE_P * 4];   // bilinear corner indices
    __shared__ float sWgt[NTAP * TILE_P * 4];   // bilinear weights (validity-masked)

    const int tid      = threadIdx.x;
    const int p_global = blockIdx.x * TILE_P;       // global pixel id in [0, 4*HW)
    const int n        = p_global >> 14;            // / HW (16384)
    const int p_img    = p_global & (HW - 1);       // pixel offset within image

    // ---- Phase 1: bilinear coefficients per (tap, pixel); reused by all 64 channels
    const float* offn = offset + (size_t)n * 18 * HW;
    for (int t = tid; t < NTAP * TILE_P; t += 256) {
        const int j  = t >> 6;          // tap 0..8
        const int pl = t & 63;          // local pixel
        const int p  = p_img + pl;
        const int ho = p >> 7;          // / W_IN
        const int wo = p & 127;
        const float dy = offn[(2 * j + 0) * HW + p];
        const float dx = offn[(2 * j + 1) * HW + p];
        const float py = (float)(ho - 1 + (j / 3)) + dy;   // stride=1, pad=1, dil=1
        const float px = (float)(wo - 1 + (j % 3)) + dx;
        const float y0f = floorf(py);
        const float x0f = floorf(px);
        const float ly = py - y0f;
        const float lx = px - x0f;
        const int y0 = (int)y0f, x0 = (int)x0f;
        const int y1 = y0 + 1,   x1 = x0 + 1;
        const bool vy0 = (y0 >= 0) && (y0 < H_IN);
        const bool vy1 = (y1 >= 0) && (y1 < H_IN);
        const bool vx0 = (x0 >= 0) && (x0 < W_IN);
        const bool vx1 = (x1 >= 0) && (x1 < W_IN);
        const int cy0 = min(max(y0, 0), H_IN - 1);
        const int cy1 = min(max(y1, 0), H_IN - 1);
        const int cx0 = min(max(x0, 0), W_IN - 1);
        const int cx1 = min(max(x1, 0), W_IN - 1);
        const int b = t * 4;
        sIdx[b + 0] = cy0 * W_IN + cx0;
        sIdx[b + 1] = cy0 * W_IN + cx1;
        sIdx[b + 2] = cy1 * W_IN + cx0;
        sIdx[b + 3] = cy1 * W_IN + cx1;
        sWgt[b + 0] = (vy0 && vx0) ? (1.f - ly) * (1.f - lx) : 0.f;
        sWgt[b + 1] = (vy0 && vx1) ? (1.f - ly) * lx         : 0.f;
        sWgt[b + 2] = (vy1 && vx0) ? ly * (1.f - lx)         : 0.f;
        sWgt[b + 3] = (vy1 && vx1) ? ly * lx                 : 0.f;
    }

    // Wave -> tile mapping: 8 waves, tile grid 4(M) x 4(N) of 16x16
    const int lane = tid & 31;
    const int wave = tid >> 5;
    const int mt   = wave >> 1;          // M tile 0..3 (out-channel block)
    const int nt0  = (wave & 1) * 2;     // first of two N tiles (pixel block)
    const int lm   = lane & 15;
    const int hk   = lane >> 4;          // ISA 16x16x4 f32 layout: K = vgpr + 2*(lane/16)
    const int pl   = tid & 63;           // sampling pixel
    const int cl   = tid >> 6;           // sampling channel-in-chunk 0..3
    const int kb   = cl * NTAP;          // k base within chunk for this thread

    v8f acc0 = {0.f, 0.f, 0.f, 0.f, 0.f, 0.f, 0.f, 0.f};
    v8f acc1 = {0.f, 0.f, 0.f, 0.f, 0.f, 0.f, 0.f, 0.f};

    const float* xn = x + (size_t)n * C_IN * HW;

    for (int c0 = 0; c0 < C_IN; c0 += CPC) {
        __syncthreads();  // prior WMMA fragment reads done; also fences phase-1 writes

        // Stage weight chunk as K-pairs. t-major mapping -> conflict-free LDS
        // stores; weight is tiny and L2-resident, so global coalescing is moot.
        for (int idx = tid; idx < KP * OC; idx += 256) {      // 1152 float2's
            const int t = idx >> 6;      // pair row 0..17
            const int m = idx & 63;      // out channel
            const v2f w2 = *(const v2f*)&weight[(size_t)m * KTOT + c0 * NTAP + 2 * t];
            *(v2f*)&sW[t * P2F + 2 * m] = w2;
        }

        // Build im2col chunk: each thread = (channel c0+cl, pixel pl), 9 taps
        const float* xc = xn + (size_t)(c0 + cl) * HW;
        #pragma unroll
        for (int j = 0; j < NTAP; ++j) {
            const int b = (j * TILE_P + pl) * 4;
            const int   i0 = sIdx[b + 0], i1 = sIdx[b + 1], i2 = sIdx[b + 2], i3 = sIdx[b + 3];
            const float w0 = sWgt[b + 0], w1 = sWgt[b + 1], w2 = sWgt[b + 2], w3 = sWgt[b + 3];
            const float s = w0 * xc[i0] + w1 * xc[i1] + w2 * xc[i2] + w3 * xc[i3];
            const int k = kb + j;
            sS[(k >> 1) * P2F + pl * 2 + (k & 1)] = s;
        }
        __syncthreads();

        // 9 f32 WMMA K-steps (K=4 each); 3 ds_load_b64 + 2 WMMA per step
        #pragma unroll
        for (int ks = 0; ks < KC; ks += 4) {
            const int row = (ks >> 1) + hk;                  // K-pair row for this lane half
            const v2f a  = *(const v2f*)&sW[row * P2F + (mt * 16 + lm) * 2];
            const v2f b0 = *(const v2f*)&sS[row * P2F + (nt0 * 16 + lm) * 2];
            const v2f b1 = *(const v2f*)&sS[row * P2F + ((nt0 + 1) * 16 + lm) * 2];
            acc0 = __builtin_amdgcn_wmma_f32_16x16x4_f32(
                false, a, false, b0, (short)0, acc0, false, false);
            acc1 = __builtin_amdgcn_wmma_f32_16x16x4_f32(
                false, a, false, b1, (short)0, acc1, false, false);
        }
    }

    // ---- Epilogue: 16x16 f32 C/D layout: VGPR r, lane half h -> M = r + 8h, N = lane%16
    float* outn = out + (size_t)n * OC * HW + p_img;
    const int pn = nt0 * 16 + lm;
    #pragma unroll
    for (int r = 0; r < 8; ++r) {
        const int o = mt * 16 + r + 8 * hk;
        outn[(size_t)o * HW + pn]      = acc0[r];
        outn[(size_t)o * HW + pn + 16] = acc1[r];
    }
}

extern "C" void kernel_launch(void* const* d_in, const int* in_sizes, int n_in,
                              void* d_out, int out_size, void* d_ws, size_t ws_size,
                              hipStream_t stream) {
    const float* x      = (const float*)d_in[0];  // input_tensor
    const float* offset = (const float*)d_in[1];  // offset
    const float* weight = (const float*)d_in[2];  // weight
    float* out = (float*)d_out;

    const int total_pixels = 4 * HW;              // N * H * W
    dim3 grid(total_pixels / TILE_P);             // 1024 workgroups
    dim3 block(256);                              // 8 wave32 waves
    hipLaunchKernelGGL(deform_conv_wmma_f32, grid, block, 0, stream,
                       x, offset, weight, out);
}